// MultiUnitCluster_21397527068765
// MI455X (gfx1250) — compile-verified
//
#include <hip/hip_runtime.h>

// MultiUnitCluster forward for MI455X (gfx1250, wave32).
// Outputs (flat in d_out): y_logits[10] | act[1,000,000] | win_ind[50,000]
// Workspace (d_ws): act_active[1M] f32 | act_win[1M] f32 | hist[4096] u32 | meta[4] u32

#define NU      1000000
#define ND      32
#define NC      10
#define TOPK_N  50000
#define NBINS   4096
#define PHI_F   1.5f

typedef __attribute__((ext_vector_type(2))) float v2f;
typedef __attribute__((ext_vector_type(8))) float v8f;

// ---------------- K0: init logits / histogram / counters ----------------
__global__ void __launch_bounds__(256) muc_init_kernel(float* __restrict__ y,
                                                       unsigned* __restrict__ hist,
                                                       unsigned* __restrict__ meta) {
  int i = blockIdx.x * blockDim.x + threadIdx.x;
  if (i < NC)    y[i]    = 0.0f;
  if (i < NBINS) hist[i] = 0u;
  if (i < 4)     meta[i] = 0u;
}

// ---------------- K1: distance + activation (WMMA dot) + histogram ----------------
// d2(u) = Sx - 2*sum_d(attn_d * x_d * w_ud) + sum_d(attn_d * w_ud^2)
// act(u) = exp(-sqrt(max(d2,0)))   (C = 1.0, R = 2.0)
__global__ void __launch_bounds__(256) muc_dist_act_kernel(
    const float* __restrict__ x, const float* __restrict__ dist_w,
    const float* __restrict__ attn_w, const float* __restrict__ active,
    float* __restrict__ act_out, float* __restrict__ act_active,
    unsigned* __restrict__ g_hist)
{
  __shared__ unsigned s_hist[NBINS];
  for (int i = threadIdx.x; i < NBINS; i += blockDim.x) s_hist[i] = 0u;
  __syncthreads();

  const int lane = threadIdx.x & 31;
  const int hi   = lane >> 4;      // 0: lanes 0-15, 1: lanes 16-31
  const int rlan = lane & 15;      // unit-in-tile owned by this lane
  const int wave = (blockIdx.x * blockDim.x + threadIdx.x) >> 5;
  const int nwav = (gridDim.x * blockDim.x) >> 5;

  // Per-wave constants: x, attention weights, Sx = sum attn*x^2
  float xv = x[lane];
  float av = attn_w[lane];
  float sx = av * xv * xv;
  #pragma unroll
  for (int off = 16; off > 0; off >>= 1) sx += __shfl_xor(sx, off, 32);

  // B operands for V_WMMA_F32_16X16X4_F32 (broadcast attn*x along N).
  // B layout (mirrors 16x4 f32 A / 16x16 C split): V0 holds rows K=4j+2*hi, V1 rows K=4j+2*hi+1.
  float b0[8], b1[8], aw0[8], aw1[8];
  #pragma unroll
  for (int j = 0; j < 8; ++j) {
    int k0 = 4 * j + 2 * hi;
    float xk0 = __shfl(xv, k0,     32);
    float xk1 = __shfl(xv, k0 + 1, 32);
    float ak0 = __shfl(av, k0,     32);
    float ak1 = __shfl(av, k0 + 1, 32);
    aw0[j] = ak0;       aw1[j] = ak1;
    b0[j]  = ak0 * xk0; b1[j]  = ak1 * xk1;
  }

  const int NTILES = NU / 16;  // 62,500 tiles of 16 units
  for (int t = wave; t < NTILES; t += nwav) {
    const int u0 = t << 4;
    // A tile: lane holds w[u0+rlan][4j+2*hi .. +1] per K-step (16x4 f32 A layout)
    const float* wrow = dist_w + (size_t)(u0 + rlan) * ND + 2 * hi;
    if (t + nwav < NTILES)  // global_prefetch_b8 of next tile this wave will touch
      __builtin_prefetch(dist_w + (size_t)(((t + nwav) << 4) + rlan) * ND, 0, 1);

    // Batch all 8 loads first -> one clause, 8 outstanding global_load_b64.
    v2f areg[8];
    #pragma unroll
    for (int j = 0; j < 8; ++j)
      areg[j] = *(const v2f*)(wrow + 4 * j);

    v8f  c  = {0.f, 0.f, 0.f, 0.f, 0.f, 0.f, 0.f, 0.f};
    float sw_a = 0.f, sw_b = 0.f;   // two accumulators for ILP
    #pragma unroll
    for (int j = 0; j < 8; ++j) {
      v2f a = areg[j];
      sw_a = fmaf(aw0[j], a.x * a.x, sw_a);       // weighted ||w||^2 partials
      sw_b = fmaf(aw1[j], a.y * a.y, sw_b);
      v2f b; b.x = b0[j]; b.y = b1[j];
      // D(16x16) += A(16x4) * B(4x16); B broadcast along N -> every column = dot
      c = __builtin_amdgcn_wmma_f32_16x16x4_f32(false, a, false, b,
                                                (short)0, c, false, false);
    }
    float sw = sw_a + sw_b;
    sw += __shfl_xor(sw, 16, 32);                 // combine the two 16-dim halves

    // Extract dot(unit = lane&15): c[v] holds unit v (lanes 0-15) / v+8 (lanes 16-31)
    float r0 = (lane & 1) ? c[1] : c[0];
    float r1 = (lane & 1) ? c[3] : c[2];
    float r2 = (lane & 1) ? c[5] : c[4];
    float r3 = (lane & 1) ? c[7] : c[6];
    float s0v = (lane & 2) ? r1 : r0;
    float s1v = (lane & 2) ? r3 : r2;
    float dsel = (lane & 4) ? s1v : s0v;          // lane holds c[lane&7]
    int   src  = (lane & 7) | ((lane & 8) << 1);  // fetch unit (lane&15)'s value
    float dot  = __shfl(dsel, src, 32);           // ds_bpermute / v_permlane

    float d2 = sx - 2.0f * dot + sw;
    d2 = d2 > 0.f ? d2 : 0.f;
    float actv = __expf(-__fsqrt_rn(d2));
    float aa   = actv * active[u0 + rlan];

    unsigned bin = __float_as_uint(aa) >> 19;     // monotone for aa in [0,1]
    if (bin >= NBINS) bin = NBINS - 1;
    if (lane < 16) {
      act_out[u0 + lane]    = actv;
      act_active[u0 + lane] = aa;
      atomicAdd(&s_hist[bin], 1u);                // ds_add_u32
    }
  }
  __syncthreads();
  for (int i = threadIdx.x; i < NBINS; i += blockDim.x) {
    unsigned v = s_hist[i];
    if (v) atomicAdd(&g_hist[i], v);
  }
}

// ---------------- K2: suffix scan of histogram -> threshold bin + tie quota ----------------
__global__ void muc_scan_kernel(const unsigned* __restrict__ hist,
                                unsigned* __restrict__ meta) {
  if (blockIdx.x == 0 && threadIdx.x == 0) {
    unsigned cum = 0, tb = 0, quota = TOPK_N;
    for (int b = NBINS - 1; b >= 0; --b) {
      unsigned cnt = hist[b];
      if (cum + cnt >= TOPK_N) { tb = (unsigned)b; quota = TOPK_N - cum; break; }
      cum += cnt;
    }
    meta[0] = tb;
    meta[1] = quota;
  }
}

// ---------------- K3: winner selection -> act_win + win_ind ----------------
__global__ void __launch_bounds__(256) muc_select_kernel(
    const float* __restrict__ act_active, float* __restrict__ act_win,
    int* __restrict__ win_ind, unsigned* __restrict__ meta) {
  int u = blockIdx.x * blockDim.x + threadIdx.x;
  if (u >= NU) return;
  float aa = act_active[u];
  unsigned bin = __float_as_uint(aa) >> 19;
  if (bin >= NBINS) bin = NBINS - 1;
  unsigned tb = meta[0];
  bool win = false;
  if (bin > tb) {
    win = true;
  } else if (bin == tb) {
    unsigned pos = atomicAdd(&meta[2], 1u);       // tie quota
    if (pos < meta[1]) win = true;
  }
  if (win) {
    act_win[u] = aa;
    unsigned slot = atomicAdd(&meta[3], 1u);
    if (slot < TOPK_N) win_ind[slot] = u;
  } else {
    act_win[u] = 0.0f;
  }
}

// ---------------- K4: y = PHI * cls_w @ act_win (float4-vectorized GEMV) ----------------
__global__ void __launch_bounds__(256) muc_gemv_kernel(
    const float* __restrict__ cls_w, const float* __restrict__ act_win,
    float* __restrict__ y) {
  float acc[NC] = {};
  const size_t stride = (size_t)gridDim.x * blockDim.x;
  const size_t NG = NU / 4;                       // groups of 4 units, b128 loads
  for (size_t g = (size_t)blockIdx.x * blockDim.x + threadIdx.x; g < NG; g += stride) {
    float4 a = ((const float4*)act_win)[g];
    bool any = (a.x != 0.f) | (a.y != 0.f) | (a.z != 0.f) | (a.w != 0.f);
    if (__ballot(any)) {                          // skip cls_w reads for all-loser tiles
      #pragma unroll
      for (int cc = 0; cc < NC; ++cc) {
        float4 w = ((const float4*)(cls_w + (size_t)cc * NU))[g];  // global_load_b128
        acc[cc] = fmaf(w.x, a.x, acc[cc]);
        acc[cc] = fmaf(w.y, a.y, acc[cc]);
        acc[cc] = fmaf(w.z, a.z, acc[cc]);
        acc[cc] = fmaf(w.w, a.w, acc[cc]);
      }
    }
  }
  #pragma unroll
  for (int off = 16; off > 0; off >>= 1)
    #pragma unroll
    for (int cc = 0; cc < NC; ++cc) acc[cc] += __shfl_xor(acc[cc], off, 32);

  __shared__ float sacc[8 * NC];                  // 256 threads = 8 waves
  int lane = threadIdx.x & 31, w = threadIdx.x >> 5;
  if (lane == 0)
    for (int cc = 0; cc < NC; ++cc) sacc[w * NC + cc] = acc[cc];
  __syncthreads();
  if (threadIdx.x == 0) {
    int nw = blockDim.x >> 5;
    for (int cc = 0; cc < NC; ++cc) {
      float tot = 0.f;
      for (int ww = 0; ww < nw; ++ww) tot += sacc[ww * NC + cc];
      atomicAdd(&y[cc], PHI_F * tot);             // global_atomic_add_f32
    }
  }
}

extern "C" void kernel_launch(void* const* d_in, const int* in_sizes, int n_in,
                              void* d_out, int out_size, void* d_ws, size_t ws_size,
                              hipStream_t stream) {
  const float* x      = (const float*)d_in[0];   // [32]
  const float* dist_w = (const float*)d_in[1];   // [1M, 32]
  const float* attn_w = (const float*)d_in[2];   // [32]
  const float* cls_w  = (const float*)d_in[3];   // [10, 1M]
  const float* active = (const float*)d_in[4];   // [1M]

  float* out      = (float*)d_out;
  float* y        = out;                          // [10]
  float* act_out  = out + NC;                     // [1M]
  int*   win_ind  = (int*)(out + NC + NU);        // [50000]

  float*    act_active = (float*)d_ws;            // [1M]
  float*    act_win    = act_active + NU;         // [1M]
  unsigned* hist       = (unsigned*)(act_win + NU);
  unsigned* meta       = hist + NBINS;            // [0]=thresh bin [1]=quota [2]=tie ctr [3]=slot ctr

  muc_init_kernel<<<16, 256, 0, stream>>>(y, hist, meta);
  muc_dist_act_kernel<<<2048, 256, 0, stream>>>(x, dist_w, attn_w, active,
                                                act_out, act_active, hist);
  muc_scan_kernel<<<1, 32, 0, stream>>>(hist, meta);
  muc_select_kernel<<<(NU + 255) / 256, 256, 0, stream>>>(act_active, act_win,
                                                          win_ind, meta);
  muc_gemv_kernel<<<1024, 256, 0, stream>>>(cls_w, act_win, y);
}